// WLSEntryLayer_49065706389961
// MI455X (gfx1250) — compile-verified
//
#include <hip/hip_runtime.h>
#include <stdint.h>

// --------------------------------------------------------------------------
// WLSEntryLayer: polynomial feature expansion + edge-weighted gather/scatter.
//   out[dst_e, :] += w_e * [x, (x^2-1)/sqrt(2), x^3/sqrt(15)] (x = features[src_e])
// Memory-bound (L2 atomics). CDNA5 paths used: TDM tensor_load_to_lds staging
// of edge arrays into LDS (TENSORcnt), global_atomic_add_f32 scatter.
// --------------------------------------------------------------------------

typedef __attribute__((ext_vector_type(4))) unsigned int v4u;
typedef __attribute__((ext_vector_type(8))) int          v8i;
typedef __attribute__((ext_vector_type(4))) int          v4i;

#define CHUNK 256   // edges per workgroup

// Build a 1-D TDM descriptor and issue TENSOR_LOAD_TO_LDS.
// lds_off    : byte offset of destination inside this workgroup's LDS
// gptr       : global address of tile start (element 0 of this chunk)
// n_remaining: elements left in the source array from gptr (OOB -> zero fill)
// tile_elems : elements to move (4-byte elements)
__device__ __forceinline__ void tdm_load_1d_b32(uint32_t lds_off, const void* gptr,
                                                uint32_t n_remaining, uint32_t tile_elems) {
  uint64_t ga = (uint64_t)(uintptr_t)gptr;

  // ---- D# group 0 (128b): count / lds_addr / global_addr / type ----
  v4u g0;
  g0.x = 1u;                                              // count=1, is_restore=0, gather off
  g0.y = lds_off;                                         // LDS byte address
  g0.z = (uint32_t)(ga & 0xFFFFFFFFu);                    // global_addr[31:0]
  g0.w = (uint32_t)((ga >> 32) & 0x01FFFFFFu)             // global_addr[56:32]
       | (2u << 30);                                      // type = 2 ("image")

  // ---- D# group 1 (256b) ----
  uint32_t dim0 = n_remaining;                            // tensor_dim0 (elements)
  v8i g1;
  g1[0] = (int)(2u << 16);                                // workgroup_mask=0, data_size=2 (4B)
  g1[1] = (int)((dim0 & 0xFFFFu) << 16);                  // tensor_dim0[15:0] @ bits 63:48
  g1[2] = (int)(((dim0 >> 16) & 0xFFFFu)                  // tensor_dim0[31:16]
       | (1u << 16));                                     // tensor_dim1 = 1 (low 16)
  g1[3] = (int)(tile_elems << 16);                        // tensor_dim1 hi=0, tile_dim0
  g1[4] = (int)1u;                                        // tile_dim1=1, tile_dim2=0
  g1[5] = (int)dim0;                                      // tensor_dim0_stride[31:0]
  g1[6] = 0;                                              // stride0 hi, stride1 lo
  g1[7] = 0;                                              // stride1 hi

  v4i zz4 = {0, 0, 0, 0};                                 // groups 2/3 unused (<=2D)
  v8i zz8 = {0, 0, 0, 0, 0, 0, 0, 0};                     // trailing group (unused)
  __builtin_amdgcn_tensor_load_to_lds(g0, g1, zz4, zz4, zz8, /*cpol=*/0);
}

__global__ __launch_bounds__(256)
void wls_zero_kernel(float4* __restrict__ out, int n4) {
  int i = blockIdx.x * blockDim.x + threadIdx.x;
  const float4 z = make_float4(0.f, 0.f, 0.f, 0.f);
  int stride = gridDim.x * blockDim.x;
  for (; i < n4; i += stride) out[i] = z;
}

__global__ __launch_bounds__(256)
void wls_edge_kernel(const float* __restrict__ features,
                     const float* __restrict__ edge_w,
                     const int*   __restrict__ src,
                     const int*   __restrict__ dst,
                     float*       __restrict__ out,
                     int n_edges) {
  __shared__ int   s_src[CHUNK];
  __shared__ int   s_dst[CHUNK];
  __shared__ float s_w[CHUNK];

  const int base = blockIdx.x * CHUNK;
  const uint32_t remaining = (uint32_t)(n_edges - base);  // grid sized so > 0

  // Wave 0 issues the three TDM copies and waits for TENSORcnt==0.
  if (threadIdx.x == 0) {
    tdm_load_1d_b32((uint32_t)(uintptr_t)s_src, src    + base, remaining, CHUNK);
    tdm_load_1d_b32((uint32_t)(uintptr_t)s_dst, dst    + base, remaining, CHUNK);
    tdm_load_1d_b32((uint32_t)(uintptr_t)s_w,   edge_w + base, remaining, CHUNK);
    __builtin_amdgcn_s_wait_tensorcnt(0);
  }
  __syncthreads();

  const float inv_sqrt2  = 0.70710678118654752f;  // 1/sqrt(2)  (std of x^2-1)
  const float inv_sqrt15 = 0.25819888974716110f;  // 1/sqrt(15) (std of x^3)

  // 256 edges * 16 dims = 4096 (edge,dim) pairs; 16 per thread.
  // dim = id & 15 -> coalesced 64B feature reads and coalesced atomic groups.
#pragma unroll
  for (int it = 0; it < 16; ++it) {
    int li = it * 256 + (int)threadIdx.x;
    int el = li >> 4;        // local edge 0..255
    int d  = li & 15;        // channel   0..15
    int e  = base + el;
    if (e < n_edges) {
      int   s = s_src[el];
      int   t = s_dst[el];
      float w = s_w[el];
      float x  = features[s * 16 + d];
      float x2 = x * x;
      float m1 = w * x;
      float m2 = w * ((x2 - 1.0f) * inv_sqrt2);
      float m3 = w * (x2 * x * inv_sqrt15);
      float* o = out + (size_t)t * 48 + d;
      atomicAdd(o,      m1);   // global_atomic_add_f32 (L2-resident output)
      atomicAdd(o + 16, m2);
      atomicAdd(o + 32, m3);
    }
  }
}

extern "C" void kernel_launch(void* const* d_in, const int* in_sizes, int n_in,
                              void* d_out, int out_size, void* d_ws, size_t ws_size,
                              hipStream_t stream) {
  const float* features = (const float*)d_in[0];   // [N,16] f32
  const float* edge_w   = (const float*)d_in[1];   // [E]    f32
  const int*   src      = (const int*)d_in[2];     // [E]    i32
  const int*   dst      = (const int*)d_in[3];     // [E]    i32
  float*       out      = (float*)d_out;           // [N,48] f32

  const int E  = in_sizes[1];
  const int n4 = out_size / 4;                     // out_size divisible by 4 (48 per node)

  wls_zero_kernel<<<512, 256, 0, stream>>>((float4*)out, n4);

  const int nblk = (E + CHUNK - 1) / CHUNK;
  wls_edge_kernel<<<nblk, 256, 0, stream>>>(features, edge_w, src, dst, out, E);
}